// VectorQuantizer_74113955660427
// MI455X (gfx1250) — compile-verified
//
#include <hip/hip_runtime.h>
#include <hip/hip_bf16.h>
#include <math.h>

typedef float v2f __attribute__((ext_vector_type(2)));
typedef float v8f __attribute__((ext_vector_type(8)));

#define NUM_CODES 1024
#define DIM       64
#define NVEC      65536      // 64*32*32
#define HW        1024       // 32*32
#define TOTAL_ELEMS 4194304  // 64*64*32*32
#define WAVES_PER_BLOCK 8

// ---------------------------------------------------------------------------
// Kernel 0: codebook norms + zero scratch (must re-zero every call: harness
// does not re-poison between graph replays).
// ---------------------------------------------------------------------------
__global__ __launch_bounds__(256) void vq_init(const float* __restrict__ cb,
                                               float* __restrict__ cnorm,
                                               unsigned* __restrict__ ucnt,
                                               float* __restrict__ ssq) {
  int k = blockIdx.x * 256 + threadIdx.x;   // 0..1023
  if (k < NUM_CODES) {
    const float4* p = (const float4*)(cb + k * DIM);
    float s = 0.f;
#pragma unroll
    for (int i = 0; i < DIM / 4; ++i) {
      float4 v = p[i];
      s += v.x * v.x + v.y * v.y + v.z * v.z + v.w * v.w;
    }
    cnorm[k] = s;
    ucnt[k] = 0u;
    if (k == 0) ssq[0] = 0.f;
  }
}

// ---------------------------------------------------------------------------
// Kernel 1: per-wave 16-row tile; f32 WMMA score = z.c - 0.5*||c||^2 over all
// 1024 codes. Software-pipelined B fragments (next tile loads overlap current
// tile's WMMAs) and dual independent accumulator chains for WMMA ILP.
// ---------------------------------------------------------------------------
__global__ __launch_bounds__(256) void vq_main(const float* __restrict__ ze,
                                               const float* __restrict__ cb,
                                               const float* __restrict__ cnorm,
                                               float* __restrict__ zq_out,
                                               float* __restrict__ idx_out,
                                               unsigned* __restrict__ ucnt,
                                               float* __restrict__ ssq) {
  __shared__ int lidx[WAVES_PER_BLOCK][16];

  const int lane = threadIdx.x & 31;
  const int wave = threadIdx.x >> 5;
  const int tile = blockIdx.x * WAVES_PER_BLOCK + wave;  // 0..4095
  const int b    = tile >> 6;                            // 16 rows share b
  const int hw0  = (tile & 63) * 16;

  const float* zbase = ze + b * (DIM * HW) + hw0;        // [k*HW + m]
  const int m     = lane & 15;                           // A row
  const int khalf = (lane >> 4) * 2;                     // K sub-offset 0|2
  const int ncol  = lane & 15;                           // code column in tile

  // --- A fragments: 16 x (K=4) f32 fragments covering DIM=64 ---
  v2f a[16];
#pragma unroll
  for (int f = 0; f < 16; ++f) {
    int k = f * 4 + khalf;
    a[f][0] = zbase[k * HW + m];
    a[f][1] = zbase[(k + 1) * HW + m];
  }

  float best[8];
  int   bidx[8];
#pragma unroll
  for (int r = 0; r < 8; ++r) { best[r] = -1e30f; bidx[r] = 0; }

  // --- software-pipelined sweep over all 1024 codes, 16 at a time ---
  v2f  bcur[16];
  float cncur;
  {
    const float* crow = cb + ncol * DIM + khalf;
#pragma unroll
    for (int f = 0; f < 16; ++f) bcur[f] = *(const v2f*)(crow + f * 4);
    cncur = cnorm[ncol];
  }

  for (int t = 0; t < NUM_CODES / 16; ++t) {
    v2f  bnext[16];
    float cnnext = 0.f;
    if (t < NUM_CODES / 16 - 1) {
      const int ncode = (t + 1) * 16 + ncol;
      const float* crow = cb + ncode * DIM + khalf;
#pragma unroll
      for (int f = 0; f < 16; ++f) bnext[f] = *(const v2f*)(crow + f * 4);
      cnnext = cnorm[ncode];
    }

    const int code = t * 16 + ncol;
    const float c0 = -0.5f * cncur;
    v8f acc0 = {c0, c0, c0, c0, c0, c0, c0, c0};
    v8f acc1 = {0.f, 0.f, 0.f, 0.f, 0.f, 0.f, 0.f, 0.f};
#pragma unroll
    for (int f = 0; f < 8; ++f) {
      acc0 = __builtin_amdgcn_wmma_f32_16x16x4_f32(false, a[2 * f], false,
                                                   bcur[2 * f], (short)0, acc0,
                                                   false, false);
      acc1 = __builtin_amdgcn_wmma_f32_16x16x4_f32(false, a[2 * f + 1], false,
                                                   bcur[2 * f + 1], (short)0,
                                                   acc1, false, false);
    }
#pragma unroll
    for (int r = 0; r < 8; ++r) {
      float s = acc0[r] + acc1[r];
      if (s > best[r]) { best[r] = s; bidx[r] = code; }
    }

#pragma unroll
    for (int f = 0; f < 16; ++f) bcur[f] = bnext[f];
    cncur = cnnext;
  }

  // --- cross-lane argmax within each 16-lane half (tie -> lower index) ---
#pragma unroll
  for (int mask = 1; mask < 16; mask <<= 1) {
#pragma unroll
    for (int r = 0; r < 8; ++r) {
      float ov = __shfl_xor(best[r], mask, 32);
      int   oi = __shfl_xor(bidx[r], mask, 32);
      if (ov > best[r] || (ov == best[r] && oi < bidx[r])) {
        best[r] = ov;
        bidx[r] = oi;
      }
    }
  }

  // lanes 0 / 16 hold final results for rows 0..7 / 8..15
  if ((lane & 15) == 0) {
    const int roff = (lane >> 4) * 8;
#pragma unroll
    for (int r = 0; r < 8; ++r) {
      int id = bidx[r];
      lidx[wave][roff + r] = id;
      idx_out[tile * 16 + roff + r] = (float)id;
      atomicAdd(&ucnt[id], 1u);
    }
  }
  __syncthreads();

  // --- gather zq (== zq_st numerically) + MSE accumulation ---
  const int gm    = lane & 15;          // row handled by this lane
  const int chalf = (lane >> 4) * 32;   // channel half
  const int myidx = lidx[wave][gm];
  const float* gsrc = cb + myidx * DIM + chalf;
  const float* zsrc = zbase + gm;
  float* qdst = zq_out + b * (DIM * HW) + hw0 + gm;
  float lsum = 0.f;
#pragma unroll
  for (int c = 0; c < 32; ++c) {
    float q = gsrc[c];
    float z = zsrc[(chalf + c) * HW];
    qdst[(chalf + c) * HW] = q;
    float d = q - z;
    lsum += d * d;
  }
#pragma unroll
  for (int mask = 1; mask < 32; mask <<= 1) lsum += __shfl_xor(lsum, mask, 32);
  if (lane == 0) atomicAdd(ssq, lsum);
}

// ---------------------------------------------------------------------------
// Kernel 2: usage / perplexity / dead codes / losses
// ---------------------------------------------------------------------------
__global__ __launch_bounds__(1024) void vq_finalize(const unsigned* __restrict__ ucnt,
                                                    const float* __restrict__ ssq,
                                                    float* __restrict__ scal_out,
                                                    float* __restrict__ usage_out) {
  __shared__ float se[32], sd[32];
  const int t = threadIdx.x;           // 0..1023
  unsigned c = ucnt[t];
  float u = (float)c * (1.0f / (float)NVEC);
  usage_out[t] = u;
  float e = u * logf(u + 1e-10f);
  float d = (c < 2u) ? 1.0f : 0.0f;    // usage*n < 2.0
#pragma unroll
  for (int mask = 1; mask < 32; mask <<= 1) {
    e += __shfl_xor(e, mask, 32);
    d += __shfl_xor(d, mask, 32);
  }
  const int wv = t >> 5, ln = t & 31;
  if (ln == 0) { se[wv] = e; sd[wv] = d; }
  __syncthreads();
  if (t == 0) {
    float te = 0.f, td = 0.f;
    for (int i = 0; i < 32; ++i) { te += se[i]; td += sd[i]; }
    float loss = ssq[0] * (1.0f / (float)TOTAL_ELEMS);
    scal_out[0] = loss;          // codebook_loss
    scal_out[1] = loss;          // commitment_loss (identical numerically)
    scal_out[2] = expf(-te);     // perplexity
    scal_out[3] = td;            // dead_codes
  }
}

// ---------------------------------------------------------------------------
extern "C" void kernel_launch(void* const* d_in, const int* in_sizes, int n_in,
                              void* d_out, int out_size, void* d_ws, size_t ws_size,
                              hipStream_t stream) {
  const float* ze = (const float*)d_in[0];   // [64,64,32,32]
  const float* cb = (const float*)d_in[1];   // [1024,64]

  float* zq_out   = (float*)d_out;                        // 4194304
  float* idx_out  = zq_out + TOTAL_ELEMS;                 // 65536
  float* scal_out = idx_out + NVEC;                       // 4 scalars
  float* usage_out = scal_out + 4;                        // 1024

  float*    cnorm = (float*)d_ws;                         // 1024 f32
  unsigned* ucnt  = (unsigned*)((char*)d_ws + 4096);      // 1024 u32
  float*    ssq   = (float*)((char*)d_ws + 8192);         // 1 f32

  vq_init<<<4, 256, 0, stream>>>(cb, cnorm, ucnt, ssq);
  vq_main<<<NVEC / 16 / WAVES_PER_BLOCK, 256, 0, stream>>>(
      ze, cb, cnorm, zq_out, idx_out, ucnt, ssq);
  vq_finalize<<<1, 1024, 0, stream>>>(ucnt, ssq, scal_out, usage_out);
}